// MessagePassingConvolution_30176440222431
// MI455X (gfx1250) — compile-verified
//
#include <hip/hip_runtime.h>

// MACE-style message-passing convolution, fused single pass, MI455X (gfx1250).
// MLP runs on V_WMMA_F32_16X16X4_F32 (exact f32 matrix math), edges = N axis,
// transposed weights = M axis, activations ping-pong through per-wave LDS.

typedef float v2f __attribute__((ext_vector_type(2)));
typedef float v4f __attribute__((ext_vector_type(4)));
typedef float v8f __attribute__((ext_vector_type(8)));

#define CDIM   64
#define NCH    256     // node feature row = C + 3C
#define RAD    8
#define HID    64
#define NOUT   256
#define OUTW   512     // output row = 2C + 6C
#define R_W0   12      // padded LDS row stride for W0^T (8 in-features)
#define R_W    68      // padded LDS row stride for W1/W2/W3^T (64 in-features)
#define R_ACT  68      // padded LDS row stride for activations (<=64 features)
#define WAVES  16      // 512-thread blocks
#define PER_WAVE (2 * 16 * R_ACT + 48 + 32)   // actA, actB, Y[16][3], snd[16], rcv[16]

__device__ __forceinline__ float silu_f(float x) {
  // x * sigmoid(x), with hardware v_rcp_f32 (~1 ulp) instead of the IEEE
  // divide expansion (v_div_scale/v_div_fmas chains, ~10 VALU ops each).
  float e = __expf(-x);
  return x * __builtin_amdgcn_rcpf(1.0f + e);
}

__device__ __forceinline__ v8f wmma_f32(v2f a, v2f b, v8f c) {
  // D(16x16) = A(16x4) * B(4x16) + C, all f32.
  return __builtin_amdgcn_wmma_f32_16x16x4_f32(false, a, false, b, (short)0, c,
                                               false, false);
}

// One 64->64 SiLU layer for a 16-edge tile.
// actIn/actOut: [16 edges][R_ACT] LDS tiles; wt: [64 out][R_W] transposed weights.
// Lane mapping (wave32): e16 = lane&15 (edge / N), h = lane>>4 (K half-pair).
__device__ __forceinline__ void hidden_layer(const float* __restrict__ actIn,
                                             float* __restrict__ actOut,
                                             const float* __restrict__ wt,
                                             int e16, int h, float scale) {
  v8f acc[4] = {};
  #pragma unroll
  for (int t = 0; t < 16; ++t) {
    // B operand: B[k][edge] = actIn[edge][k], k = 4t + 2h + {0,1}
    v2f b = *(const v2f*)&actIn[e16 * R_ACT + 4 * t + 2 * h];
    #pragma unroll
    for (int m = 0; m < 4; ++m) {
      // A operand: A[out][k] = wt[16m + e16][k]
      v2f a = *(const v2f*)&wt[(16 * m + e16) * R_W + 4 * t + 2 * h];
      acc[m] = wmma_f32(a, b, acc[m]);
    }
  }
  // D layout: lane holds D[v + 8h][e16] for v=0..7 -> contiguous features.
  float* dst = &actOut[e16 * R_ACT + 8 * h];
  #pragma unroll
  for (int m = 0; m < 4; ++m) {
    v4f lo = { silu_f(acc[m][0] * scale), silu_f(acc[m][1] * scale),
               silu_f(acc[m][2] * scale), silu_f(acc[m][3] * scale) };
    v4f hi = { silu_f(acc[m][4] * scale), silu_f(acc[m][5] * scale),
               silu_f(acc[m][6] * scale), silu_f(acc[m][7] * scale) };
    *(v4f*)(dst + 16 * m)     = lo;
    *(v4f*)(dst + 16 * m + 4) = hi;
  }
}

extern "C" __global__ void __launch_bounds__(256, 4)
zero_f32v4(float* __restrict__ p, int n4) {
  v4f z = {};
  int i = blockIdx.x * blockDim.x + threadIdx.x;
  int stride = gridDim.x * blockDim.x;
  for (; i < n4; i += stride) ((v4f*)p)[i] = z;
}

extern "C" __global__ void __launch_bounds__(512, 1)
mace_mp_conv(const float* __restrict__ node_feats,
             const float* __restrict__ vectors,
             const float* __restrict__ radial,
             const int*   __restrict__ senders,
             const int*   __restrict__ receivers,
             const float* __restrict__ W0,
             const float* __restrict__ W1,
             const float* __restrict__ W2,
             const float* __restrict__ W3,
             float* __restrict__ out,
             int n_edges) {
  extern __shared__ float lds[];
  float* wt0 = lds;                       // 64 x R_W0
  float* wt1 = wt0 + 64 * R_W0;           // 64 x R_W
  float* wt2 = wt1 + HID * R_W;           // 64 x R_W
  float* wt3 = wt2 + HID * R_W;           // 256 x R_W
  float* wbase = wt3 + NOUT * R_W;

  const int tid  = threadIdx.x;
  const int wv   = tid >> 5;
  const int lane = tid & 31;
  const int e16  = lane & 15;
  const int h    = lane >> 4;

  float* actA = wbase + wv * PER_WAVE;
  float* actB = actA + 16 * R_ACT;
  float* yln  = actB + 16 * R_ACT;        // 16 x 3
  int*   snds = (int*)(yln + 48);
  int*   rcvs = snds + 16;

  // ---- stage weights transposed into LDS: wt[o][i] = W[i][o] (L2-resident) ----
  for (int idx = tid; idx < 64 * RAD; idx += blockDim.x) {
    int o = idx >> 3, i = idx & 7;
    wt0[o * R_W0 + i] = W0[i * 64 + o];
  }
  for (int idx = tid; idx < HID * HID; idx += blockDim.x) {
    int o = idx >> 6, i = idx & 63;
    wt1[o * R_W + i] = W1[i * HID + o];
    wt2[o * R_W + i] = W2[i * HID + o];
  }
  for (int idx = tid; idx < NOUT * HID; idx += blockDim.x) {
    int o = idx >> 6, i = idx & 63;
    wt3[o * R_W + i] = W3[i * NOUT + o];
  }
  __syncthreads();

  const float SQRT3     = 1.7320508075688772f;
  const float INV_SQRT3 = 0.57735026918962576f;
  const float S0 = 0.35355339059327378f;     // 1/sqrt(8)
  const float S  = 0.125f;                   // 1/sqrt(64)
  const float S3 = 0.125f * (1.0f / 16.0f);  // layer-3 scale * 1/avg_num_neighbors

  const int tiles = (n_edges + 15) >> 4;
  const int nwav  = gridDim.x * (blockDim.x >> 5);

  for (int tile = blockIdx.x * (blockDim.x >> 5) + wv; tile < tiles; tile += nwav) {
    // ---- per-tile edge metadata + radial staging (lanes 0..15; per-wave LDS,
    //      same-wave DS ops are in-order so no barrier needed) ----
    if (lane < 16) {
      int e = tile * 16 + lane;
      if (e < n_edges) {
        snds[lane] = senders[e];
        rcvs[lane] = receivers[e];
        float vx = vectors[e * 3 + 0], vy = vectors[e * 3 + 1], vz = vectors[e * 3 + 2];
        float rs = SQRT3 * rsqrtf(vx * vx + vy * vy + vz * vz);
        yln[lane * 3 + 0] = -vx * rs;
        yln[lane * 3 + 1] = -vy * rs;
        yln[lane * 3 + 2] = -vz * rs;
        #pragma unroll
        for (int k = 0; k < RAD; ++k) actA[lane * R_ACT + k] = radial[e * RAD + k];
      } else {
        snds[lane] = 0;
        rcvs[lane] = -1;
        yln[lane * 3 + 0] = 0.f; yln[lane * 3 + 1] = 0.f; yln[lane * 3 + 2] = 0.f;
        #pragma unroll
        for (int k = 0; k < RAD; ++k) actA[lane * R_ACT + k] = 0.f;
      }
    }

    // ---- layer 0: (16 edges x 8) @ (8 -> 64), K = 8 = two k-steps ----
    {
      v8f acc[4] = {};
      #pragma unroll
      for (int t = 0; t < 2; ++t) {
        v2f b = *(const v2f*)&actA[e16 * R_ACT + 4 * t + 2 * h];
        #pragma unroll
        for (int m = 0; m < 4; ++m) {
          v2f a = *(const v2f*)&wt0[(16 * m + e16) * R_W0 + 4 * t + 2 * h];
          acc[m] = wmma_f32(a, b, acc[m]);
        }
      }
      #pragma unroll
      for (int m = 0; m < 4; ++m) {
        float* dst = &actB[e16 * R_ACT + 16 * m + 8 * h];
        v4f lo = { silu_f(acc[m][0] * S0), silu_f(acc[m][1] * S0),
                   silu_f(acc[m][2] * S0), silu_f(acc[m][3] * S0) };
        v4f hi = { silu_f(acc[m][4] * S0), silu_f(acc[m][5] * S0),
                   silu_f(acc[m][6] * S0), silu_f(acc[m][7] * S0) };
        *(v4f*)dst       = lo;
        *(v4f*)(dst + 4) = hi;
      }
    }

    hidden_layer(actB, actA, wt1, e16, h, S);   // layer 1
    hidden_layer(actA, actB, wt2, e16, h, S);   // layer 2

    // ---- layer 3 (64 -> 256) fused with message build + scatter-add ----
    const int snd = snds[e16];
    const int rcv = rcvs[e16];
    const float y0 = yln[e16 * 3 + 0], y1 = yln[e16 * 3 + 1], y2 = yln[e16 * 3 + 2];
    const float* nf = node_feats + (long long)snd * NCH;
    float* orow = out + (long long)(rcv < 0 ? 0 : rcv) * OUTW;

    // Hoist B operands for the whole K sweep (16 x v2f = 32 VGPRs), reused 4x.
    v2f bq[16];
    #pragma unroll
    for (int t = 0; t < 16; ++t)
      bq[t] = *(const v2f*)&actB[e16 * R_ACT + 4 * t + 2 * h];

    #pragma unroll
    for (int g = 0; g < 4; ++g) {         // g selects mix segment: uniform branch
      v8f acc[4] = {};
      #pragma unroll
      for (int t = 0; t < 16; ++t) {
        #pragma unroll
        for (int mm = 0; mm < 4; ++mm) {
          v2f a = *(const v2f*)&wt3[(16 * (4 * g + mm) + e16) * R_W + 4 * t + 2 * h];
          acc[mm] = wmma_f32(a, bq[t], acc[mm]);
        }
      }
      if (rcv >= 0) {
        #pragma unroll
        for (int mm = 0; mm < 4; ++mm) {
          #pragma unroll
          for (int v = 0; v < 8; ++v) {
            const int o = 16 * (4 * g + mm) + v + 8 * h;   // mix index 0..255
            const float mix = acc[mm][v] * S3;             // incl. 1/16 norm
            if (g == 0) {                                  // msg_s[0:64] = s_e
              atomicAdd(&orow[o], nf[o] * mix);
            } else if (g == 1) {                           // msg_s[64:128] = tp_s
              const int c = o - 64;
              const float* ve = nf + CDIM + 3 * c;
              float tps = (ve[0] * y0 + ve[1] * y1 + ve[2] * y2) * INV_SQRT3;
              atomicAdd(&orow[o], tps * mix);
            } else if (g == 2) {                           // msg_v rows 0..63 = v_e
              const int c = o - 128;
              const float* ve = nf + CDIM + 3 * c;
              float* ob = &orow[128 + 3 * c];
              atomicAdd(ob + 0, ve[0] * mix);
              atomicAdd(ob + 1, ve[1] * mix);
              atomicAdd(ob + 2, ve[2] * mix);
            } else {                                       // msg_v rows 64..127 = tp_v
              const int c = o - 192;
              const float s = nf[c];
              float* ob = &orow[128 + 3 * (64 + c)];
              atomicAdd(ob + 0, s * y0 * mix);
              atomicAdd(ob + 1, s * y1 * mix);
              atomicAdd(ob + 2, s * y2 * mix);
            }
          }
        }
      }
    }
  }
}

extern "C" void kernel_launch(void* const* d_in, const int* in_sizes, int n_in,
                              void* d_out, int out_size, void* d_ws, size_t ws_size,
                              hipStream_t stream) {
  const float* node_feats = (const float*)d_in[0];
  const float* vectors    = (const float*)d_in[1];
  const float* radial     = (const float*)d_in[2];
  const int*   senders    = (const int*)d_in[3];
  const int*   receivers  = (const int*)d_in[4];
  const float* W0 = (const float*)d_in[5];
  const float* W1 = (const float*)d_in[6];
  const float* W2 = (const float*)d_in[7];
  const float* W3 = (const float*)d_in[8];
  float* out = (float*)d_out;
  const int n_edges = in_sizes[3];

  // Output rows are 512 f32 -> out_size divisible by 4.
  int n4 = out_size >> 2;
  int zb = (n4 + 2047) / 2048;
  if (zb > 2048) zb = 2048;
  zero_f32v4<<<zb, 256, 0, stream>>>(out, n4);

  size_t lds_bytes =
      (size_t)(64 * R_W0 + 2 * HID * R_W + NOUT * R_W + WAVES * PER_WAVE) *
      sizeof(float);   // ~246 KB, fits 320 KB/WGP
  (void)hipFuncSetAttribute(reinterpret_cast<const void*>(mace_mp_conv),
                            hipFuncAttributeMaxDynamicSharedMemorySize,
                            (int)lds_bytes);

  int tiles = (n_edges + 15) >> 4;
  int maxblocks = (tiles + WAVES - 1) / WAVES;
  int blocks = maxblocks < 256 ? maxblocks : 256;
  mace_mp_conv<<<blocks, 512, lds_bytes, stream>>>(
      node_feats, vectors, radial, senders, receivers, W0, W1, W2, W3, out,
      n_edges);
}